// GAT_15427522527698
// MI455X (gfx1250) — compile-verified
//
#include <hip/hip_runtime.h>
#include <hip/hip_bf16.h>
#include <math.h>

// ---------------------------------------------------------------------------
// GAT stack on dense block graphs (N1=N2=512, F=128).
// inside graph: dst i attends to its own 512-group (self-loop included).
// cross  graph: dst i attends to the other 512-group plus itself.
// ---------------------------------------------------------------------------

typedef float v2f __attribute__((ext_vector_type(2)));
typedef float v8f __attribute__((ext_vector_type(8)));

#define NNODES 1024
#define F 128
#define GRP 512
#define NEG_SLOPE 0.2f

// ---------------------------------------------------------------------------
// C[1024x128] = A[1024x128] * B[128x128], fp32 WMMA 16x16x4.
// One wave per 16x16 output tile; 8 waves/block; 64 blocks = 512 tiles.
// B (weights, 64KB) staged in LDS, reused by all 8 waves across 32 K-steps.
// A layout (ISA 7.12.2, 32-bit A 16x4): lanes 0-15 hold K=0,1 (VGPR0,1) for
// row M=lane; lanes 16-31 hold K=2,3. B (4x16) is the mirrored layout on N.
// D (16x16 f32): VGPR r -> row r (lanes 0-15) / row r+8 (lanes 16-31).
// ---------------------------------------------------------------------------
__global__ void __launch_bounds__(256) gemm1024x128x128_wmma_f32(
    const float* __restrict__ A, const float* __restrict__ B,
    float* __restrict__ C)
{
    __shared__ float Wlds[F * F];   // 64 KB of the 320 KB WGP LDS

    // cooperative stage of B into LDS (float4, fully coalesced)
    {
        const float4* __restrict__ Bv = (const float4*)B;
        float4* Wv = (float4*)Wlds;
        #pragma unroll
        for (int idx = threadIdx.x; idx < (F * F) / 4; idx += 256)
            Wv[idx] = Bv[idx];
    }
    __syncthreads();

    const int wave = threadIdx.x >> 5;
    const int lane = threadIdx.x & 31;
    const int half = lane >> 4;     // 0: K pair {0,1}/rows 0-7, 1: {2,3}/rows 8-15
    const int l16  = lane & 15;

    const int tile = blockIdx.x * 8 + wave;   // 0..511
    const int tm = tile >> 3;                 // 0..63  (M tile)
    const int tn = tile & 7;                  // 0..7   (N tile)

    const int arow = tm * 16 + l16;
    const int bcol = tn * 16 + l16;
    const float* __restrict__ Arow = A + arow * F + 2 * half;

    v8f acc = {};
    #pragma unroll
    for (int k = 0; k < F; k += 4) {
        v2f a, b;
        a.x = Arow[k];
        a.y = Arow[k + 1];
        b.x = Wlds[(k + 2 * half) * F + bcol];
        b.y = Wlds[(k + 2 * half + 1) * F + bcol];
        // (neg_a, A, neg_b, B, c_mod, C, reuse_a, reuse_b)
        acc = __builtin_amdgcn_wmma_f32_16x16x4_f32(
            false, a, false, b, (short)0, acc, false, false);
    }

    const int orow = tm * 16 + half * 8;
    #pragma unroll
    for (int r = 0; r < 8; ++r)
        C[(orow + r) * F + bcol] = acc[r];
}

// ---------------------------------------------------------------------------
// Multi-head (H=128, C=1) dense GAT attention + bias + ELU.
// One block per dst node i, one lane per head. Flash-style online softmax
// entirely in registers (no cross-lane traffic needed since C=1).
// graph==0: inside (own group, incl. self). graph==1: cross (other group)+self.
// ---------------------------------------------------------------------------
__device__ __forceinline__ void attn_accum(float hj, float as_h, float di,
                                           float& m, float& l, float& acc)
{
    float e = fmaf(hj, as_h, di);
    e = (e > 0.f) ? e : NEG_SLOPE * e;          // leaky_relu
    if (e > m) {                                 // rescale running state
        float s = __expf(m - e);
        l *= s; acc *= s; m = e;
    }
    float p = __expf(e - m);
    l += p;
    acc = fmaf(p, hj, acc);
}

__global__ void __launch_bounds__(128) gat_attn_h128(
    const float* __restrict__ H, const float* __restrict__ asrc,
    const float* __restrict__ adst, const float* __restrict__ bias,
    int graph, float* __restrict__ Xout)
{
    const int i  = blockIdx.x;     // dst node
    const int hd = threadIdx.x;    // head

    const float as_h = asrc[hd];
    const float di   = H[i * F + hd] * adst[hd];

    const int g    = i >> 9;                       // group of dst
    const int jbeg = (graph == 0) ? g * GRP : (1 - g) * GRP;

    float m = -3.0e38f, l = 0.f, acc = 0.f;
    for (int j = jbeg; j < jbeg + GRP; ++j) {
        __builtin_prefetch(&H[(j + 16) * F + hd], 0, 0);  // global_prefetch_b8
        attn_accum(H[j * F + hd], as_h, di, m, l, acc);
    }
    if (graph == 1)                                 // explicit self-loop
        attn_accum(H[i * F + hd], as_h, di, m, l, acc);

    float o = acc / (l + 1e-16f) + bias[hd];
    o = (o > 0.f) ? o : (__expf(o) - 1.f);          // ELU
    Xout[i * F + hd] = o;
}

// ---------------------------------------------------------------------------
// Head-1 attention scores: AS[n] = <H[n,:], a_src>, AD[n] = <H[n,:], a_dst>.
// One wave32 per node, shuffle-reduced.
// ---------------------------------------------------------------------------
__global__ void __launch_bounds__(256) gat_scores_h1(
    const float* __restrict__ H, const float* __restrict__ asrc,
    const float* __restrict__ adst, float* __restrict__ AS,
    float* __restrict__ AD)
{
    const int node = (blockIdx.x * blockDim.x + threadIdx.x) >> 5;
    const int lane = threadIdx.x & 31;
    if (node >= NNODES) return;

    const float* __restrict__ hrow = H + node * F;
    float s = 0.f, d = 0.f;
    #pragma unroll
    for (int c = lane; c < F; c += 32) {
        const float h = hrow[c];
        s = fmaf(h, asrc[c], s);
        d = fmaf(h, adst[c], d);
    }
    #pragma unroll
    for (int off = 16; off > 0; off >>= 1) {
        s += __shfl_xor(s, off, 32);
        d += __shfl_xor(d, off, 32);
    }
    if (lane == 0) { AS[node] = s; AD[node] = d; }
}

// ---------------------------------------------------------------------------
// Head-1 (C=128) cross-graph attention: out[i,:] = softmax-weighted sum of
// H rows over (other group + self), plus bias. One block per dst node,
// one lane per channel; scalar softmax state replicated per lane.
// ---------------------------------------------------------------------------
__global__ void __launch_bounds__(128) gat_attn_h1(
    const float* __restrict__ H, const float* __restrict__ AS,
    const float* __restrict__ AD, const float* __restrict__ bias,
    float* __restrict__ Out)
{
    const int i = blockIdx.x;
    const int c = threadIdx.x;

    const float adi = AD[i];
    const int g    = i >> 9;
    const int jbeg = (1 - g) * GRP;

    float m = -3.0e38f, l = 0.f, acc = 0.f;
    for (int jj = 0; jj <= GRP; ++jj) {
        const int j = (jj < GRP) ? (jbeg + jj) : i;   // neighborhood + self
        float e = AS[j] + adi;
        e = (e > 0.f) ? e : NEG_SLOPE * e;
        if (e > m) { float s = __expf(m - e); l *= s; acc *= s; m = e; }
        float p = __expf(e - m);
        l += p;
        acc = fmaf(p, H[j * F + c], acc);
    }
    Out[i * F + c] = acc / (l + 1e-16f) + bias[c];
}

// ---------------------------------------------------------------------------
// x = concat(desc1, desc2)
// ---------------------------------------------------------------------------
__global__ void __launch_bounds__(256) concat_inputs(
    const float* __restrict__ d1, const float* __restrict__ d2,
    float* __restrict__ X)
{
    const int idx = blockIdx.x * blockDim.x + threadIdx.x;  // 0..131071
    X[idx] = (idx < GRP * F) ? d1[idx] : d2[idx - GRP * F];
}

// ---------------------------------------------------------------------------
extern "C" void kernel_launch(void* const* d_in, const int* in_sizes, int n_in,
                              void* d_out, int out_size, void* d_ws, size_t ws_size,
                              hipStream_t stream)
{
    const float* desc1 = (const float*)d_in[0];
    const float* desc2 = (const float*)d_in[1];
    const float* W1    = (const float*)d_in[2];
    const float* asrc1 = (const float*)d_in[3];
    const float* adst1 = (const float*)d_in[4];
    const float* b1    = (const float*)d_in[5];
    const float* W2    = (const float*)d_in[6];
    const float* asrc2 = (const float*)d_in[7];
    const float* adst2 = (const float*)d_in[8];
    const float* b2    = (const float*)d_in[9];
    // d_in[10]/d_in[11] (edge lists) unused: graphs are dense by construction.

    float* X  = (float*)d_ws;              // [1024 x 128]
    float* H  = X + NNODES * F;            // [1024 x 128]
    float* AS = H + NNODES * F;            // [1024]
    float* AD = AS + NNODES;               // [1024]
    float* out = (float*)d_out;            // [1024 x 128]

    concat_inputs<<<(NNODES * F) / 256, 256, 0, stream>>>(desc1, desc2, X);

    // 4 GAT(H=128,C=1) layers: inside, cross, inside, cross — all share W1.
    for (int layer = 0; layer < 4; ++layer) {
        gemm1024x128x128_wmma_f32<<<64, 256, 0, stream>>>(X, W1, H);
        gat_attn_h128<<<NNODES, 128, 0, stream>>>(H, asrc1, adst1, b1,
                                                  layer & 1, X);
    }

    // Final GAT(H=1,C=128) on the cross graph, no ELU.
    gemm1024x128x128_wmma_f32<<<64, 256, 0, stream>>>(X, W2, H);
    gat_scores_h1<<<NNODES / 8, 256, 0, stream>>>(H, asrc2, adst2, AS, AD);
    gat_attn_h1<<<NNODES, 128, 0, stream>>>(H, AS, AD, b2, out);
}